// ShapeletLayer_88064009437909
// MI455X (gfx1250) — compile-verified
//
#include <hip/hip_runtime.h>
#include <math.h>

// Shapelet min-distance layer for MI455X (gfx1250), fp32 WMMA path.
// B=64, T=8192, K=64, L=64, n_win=8129, out = (64,64) fp32.

typedef __attribute__((ext_vector_type(2))) float v2f;
typedef __attribute__((ext_vector_type(8))) float v8f;

#define B_ 64
#define T_ 8192
#define K_ 64
#define L_ 64
#define NWIN (T_ - L_ + 1)                 // 8129
#define WT 128                             // windows per workgroup
#define NTILES ((NWIN + WT - 1) / WT)      // 64
#define EPS_ 1e-8f
#define SZS 65                             // LDS stride for shapelet matrix (bank-conflict pad)

__global__ void shp_init_out(unsigned int* __restrict__ out, int n) {
    int i = blockIdx.x * blockDim.x + threadIdx.x;
    if (i < n) out[i] = 0x7F800000u;       // +inf bits
}

__global__ __launch_bounds__(256) void shapelet_min_kernel(
    const float* __restrict__ x,           // (B, T)
    const float* __restrict__ sh,          // (K, L)
    unsigned int* __restrict__ out)        // (B, K) float bits, pre-set to +inf
{
    __shared__ float sz[K_ * SZS];         // z-normed shapelets, row stride 65
    __shared__ float s2s[K_];              // sum(sz^2) per shapelet
    __shared__ float xs[WT + L_];          // x slice for this tile (192 floats)
    __shared__ float invS[WT];             // 1/(std+eps) per window
    __shared__ float p2S[WT];              // sum(pz^2) per window (+inf if invalid)
    __shared__ unsigned int minsh[K_];     // per-block min (uint-ordered float bits)

    const int tid  = threadIdx.x;
    const int b    = blockIdx.x / NTILES;
    const int tile = blockIdx.x % NTILES;
    const int w0   = tile * WT;

    // --- 1. z-normalize shapelets into LDS (threads 0..63, one row each) ---
    if (tid < K_) {
        const float* row = sh + tid * L_;
        float s1 = 0.0f, sq = 0.0f;
        #pragma unroll 8
        for (int l = 0; l < L_; ++l) { float v = row[l]; s1 += v; sq += v * v; }
        float mu  = s1 * (1.0f / L_);
        float var = fmaxf((sq - s1 * mu) * (1.0f / (L_ - 1)), 0.0f);
        float inv = 1.0f / (sqrtf(var) + EPS_);
        float z2  = 0.0f;
        #pragma unroll 8
        for (int l = 0; l < L_; ++l) {
            float z = (row[l] - mu) * inv;
            sz[tid * SZS + l] = z;
            z2 += z * z;
        }
        s2s[tid]   = z2;
        minsh[tid] = 0x7F800000u;
    }

    // --- 2. load x slice for this window tile ---
    for (int j = tid; j < WT + L_; j += 256) {
        int g = w0 + j;
        xs[j] = (g < T_) ? x[b * T_ + g] : 0.0f;
    }
    __syncthreads();

    // --- 3. per-window stats (threads 0..127, one window each) ---
    if (tid < WT) {
        if (w0 + tid < NWIN) {
            float s1 = 0.0f, sq = 0.0f;
            #pragma unroll 8
            for (int l = 0; l < L_; ++l) { float v = xs[tid + l]; s1 += v; sq += v * v; }
            float mu  = s1 * (1.0f / L_);
            float var = fmaxf((sq - s1 * mu) * (1.0f / (L_ - 1)), 0.0f);
            float inv = 1.0f / (sqrtf(var) + EPS_);
            invS[tid] = inv;
            p2S[tid]  = (float)(L_ - 1) * var * inv * inv;
        } else {
            invS[tid] = 0.0f;
            p2S[tid]  = __builtin_inff();  // invalid window -> dist = inf
        }
    }
    __syncthreads();

    // --- 4. WMMA: raw[w,k] = sum_l xs[w+l] * sz[k,l], f32 16x16x4 ---
    // Wave wv owns windows [wv*16, wv*16+16), all 64 shapelet columns.
    // A(16x4 f32): lane m = row, VGPR i + half h -> K = 2h+i
    // B(4x16 f32): lane m = col, VGPR i + half h -> K = 2h+i
    const int wv   = tid >> 5;
    const int lane = tid & 31;
    const int m    = lane & 15;
    const int h    = lane >> 4;
    const int arow = wv * 16 + m;

    v8f acc[4] = {v8f{}, v8f{}, v8f{}, v8f{}};

    // Hoist all A operands (reused across 4 column tiles): 16 steps x 2 VGPRs.
    v2f areg[16];
    #pragma unroll
    for (int s = 0; s < 16; ++s) {
        const int off = 4 * s + 2 * h;
        areg[s].x = xs[arow + off];
        areg[s].y = xs[arow + off + 1];
    }

    // Software-pipelined B: prefetch step s+1 before issuing step s WMMAs.
    v2f bcur[4], bnxt[4];
    #pragma unroll
    for (int t = 0; t < 4; ++t) {
        const int col = t * 16 + m;
        bcur[t].x = sz[col * SZS + 2 * h];
        bcur[t].y = sz[col * SZS + 2 * h + 1];
    }
    #pragma unroll
    for (int s = 0; s < 16; ++s) {
        if (s < 15) {
            const int off = 4 * (s + 1) + 2 * h;
            #pragma unroll
            for (int t = 0; t < 4; ++t) {
                const int col = t * 16 + m;
                bnxt[t].x = sz[col * SZS + off];
                bnxt[t].y = sz[col * SZS + off + 1];
            }
        }
        #pragma unroll
        for (int t = 0; t < 4; ++t)
            acc[t] = __builtin_amdgcn_wmma_f32_16x16x4_f32(
                false, areg[s], false, bcur[t], (short)0, acc[t], false, false);
        #pragma unroll
        for (int t = 0; t < 4; ++t) bcur[t] = bnxt[t];
    }

    // --- 5. epilogue: distance + min reduction ---
    float s2k[4];
    #pragma unroll
    for (int t = 0; t < 4; ++t) s2k[t] = s2s[t * 16 + m];

    #pragma unroll
    for (int t = 0; t < 4; ++t) {
        float tmin = __builtin_inff();
        #pragma unroll
        for (int v = 0; v < 8; ++v) {
            const int wl = wv * 16 + v + 8 * h;   // C/D: M = v + 8h
            float inv = invS[wl];
            float p2  = p2S[wl];
            float d2  = (p2 + s2k[t] - 2.0f * acc[t][v] * inv) * (1.0f / L_);
            float dst = sqrtf(fmaxf(d2, 0.0f) + EPS_);
            tmin = fminf(tmin, dst);
        }
        tmin = fminf(tmin, __shfl_xor(tmin, 16, 32));
        if (h == 0)
            atomicMin(&minsh[t * 16 + m], __float_as_uint(tmin));
    }
    __syncthreads();

    // --- 6. one global atomic per (b,k) per block ---
    if (tid < K_)
        atomicMin(out + b * K_ + tid, minsh[tid]);
}

extern "C" void kernel_launch(void* const* d_in, const int* in_sizes, int n_in,
                              void* d_out, int out_size, void* d_ws, size_t ws_size,
                              hipStream_t stream) {
    (void)in_sizes; (void)n_in; (void)d_ws; (void)ws_size;
    const float* x  = (const float*)d_in[0];
    const float* sh = (const float*)d_in[1];
    unsigned int* out = (unsigned int*)d_out;

    shp_init_out<<<(out_size + 255) / 256, 256, 0, stream>>>(out, out_size);
    shapelet_min_kernel<<<B_ * NTILES, 256, 0, stream>>>(x, sh, out);
}